// GraphConvolution_68418829025577
// MI455X (gfx1250) — compile-verified
//
#include <hip/hip_runtime.h>

// GraphConvolution SpMM: out = relu(segment_sum(values[:,None] * kernel[cols], rows) + bias)
// N=50000, NNZ=1.6M, UNITS=64.
//  - segment-sum: bandwidth/atomic-bound (4.5 FLOP/B) -> async global->LDS edge staging
//    (ASYNCcnt), global_prefetch_b8, wave32 readlane broadcast, L2-side global_atomic_add_f32.
//  - epilogue: bias broadcast-add mapped to the matrix pipe as a rank-1 update,
//    v_wmma_f32_16x16x4_f32 (one instruction per 16x16 output tile), relu in VALU.

#define UNITS 64
#define EDGE_TILE 256
#define BLOCK_THREADS 256
#define SPMM_MAX_BLOCKS 1024

#if __has_builtin(__builtin_amdgcn_global_load_async_to_lds_b32) && \
    __has_builtin(__builtin_amdgcn_s_wait_asynccnt)
#define USE_ASYNC_LDS 1
#else
#define USE_ASYNC_LDS 0
#endif

#if __has_builtin(__builtin_amdgcn_wmma_f32_16x16x4_f32)
#define USE_WMMA_EPILOGUE 1
#else
#define USE_WMMA_EPILOGUE 0
#endif

typedef __attribute__((ext_vector_type(2))) float v2f;
typedef __attribute__((ext_vector_type(8))) float v8f;

#if USE_ASYNC_LDS
typedef __attribute__((address_space(1))) int* gbl_i32_ptr;
typedef __attribute__((address_space(3))) int* lds_i32_ptr;

// Strip const / retype to int, then addrspace-cast generic -> global.
__device__ __forceinline__ gbl_i32_ptr as_global_b32(const void* p) {
  int* g = (int*)const_cast<void*>(p);
  return (gbl_i32_ptr)g;
}
// Addrspace-cast generic LDS pointer -> addrspace(3).
__device__ __forceinline__ lds_i32_ptr as_lds_b32(void* p) {
  return (lds_i32_ptr)(int*)p;
}
#endif

__global__ __launch_bounds__(BLOCK_THREADS) void spmm_zero_kernel(float* __restrict__ out,
                                                                  int total) {
  const int i4 = (blockIdx.x * BLOCK_THREADS + threadIdx.x) << 2;
  if (i4 + 3 < total) {
    *reinterpret_cast<float4*>(out + i4) = make_float4(0.f, 0.f, 0.f, 0.f);
  } else {
    for (int i = i4; i < total; ++i) out[i] = 0.f;
  }
}

__global__ __launch_bounds__(BLOCK_THREADS) void spmm_atomic_kernel(
    const int* __restrict__ rows, const int* __restrict__ cols,
    const float* __restrict__ vals, const float* __restrict__ kern,
    float* __restrict__ out, int nnz, int num_tiles) {
  __shared__ int s_rows[2][EDGE_TILE];
  __shared__ int s_cols[2][EDGE_TILE];
  __shared__ float s_vals[2][EDGE_TILE];

  const int tid = threadIdx.x;
  const int lane = tid & 31;
  const int wv = tid >> 5;  // 8 waves per block (wave32)

  // Stage one 256-edge tile (rows/cols/vals) into LDS buffer b.
  auto stage = [&](int t, int b) {
    const int e = t * EDGE_TILE + tid;
    if (e < nnz) {
#if USE_ASYNC_LDS
      // gfx1250 async copy: tracked on ASYNCcnt, no VGPR data movement.
      __builtin_amdgcn_global_load_async_to_lds_b32(as_global_b32(rows + e),
                                                    as_lds_b32(&s_rows[b][tid]), 0, 0);
      __builtin_amdgcn_global_load_async_to_lds_b32(as_global_b32(cols + e),
                                                    as_lds_b32(&s_cols[b][tid]), 0, 0);
      __builtin_amdgcn_global_load_async_to_lds_b32(as_global_b32(vals + e),
                                                    as_lds_b32(&s_vals[b][tid]), 0, 0);
#else
      s_rows[b][tid] = rows[e];
      s_cols[b][tid] = cols[e];
      s_vals[b][tid] = vals[e];
#endif
    } else {
      // Tail padding: val==0 makes the contribution exactly zero.
      s_rows[b][tid] = 0;
      s_cols[b][tid] = 0;
      s_vals[b][tid] = 0.f;
    }
  };

  auto wait_stage = [&]() {
#if USE_ASYNC_LDS
    __builtin_amdgcn_s_wait_asynccnt(0);  // s_wait_asynccnt 0
#endif
    __syncthreads();  // make this wave's async LDS writes visible to all waves
  };

  int tile = blockIdx.x;  // grid is capped at num_tiles, so tile < num_tiles here
  stage(tile, 0);
  int buf = 0;

  for (; tile < num_tiles; tile += gridDim.x) {
    // Wait for the tile staged last iteration (or the prologue stage).
    // This barrier also guarantees every wave finished READING buffer `buf`
    // in the previous iteration before anyone re-stages it below.
    wait_stage();

    const int nxt = tile + gridDim.x;
    if (nxt < num_tiles) stage(nxt, buf ^ 1);  // fetch next tile while computing

    // Each wave owns 32 of the 256 staged edges; lane l carries edge (wv*32+l).
    const int idx = (wv << 5) + lane;
    const int r = s_rows[buf][idx];
    const int c = s_cols[buf][idx];
    const float v = s_vals[buf][idx];
    const int f2 = lane << 1;  // this lane's feature pair (2 floats = 8B)

    // Hint the upcoming gather rows into cache (global_prefetch_b8).
    __builtin_prefetch(kern + c * UNITS + f2, 0, 1);

#pragma unroll
    for (int i = 0; i < 32; ++i) {
      // Broadcast edge i's scalars to SGPRs -> uniform base addresses,
      // so the 32-lane gather/scatter below is one coalesced 256B access.
      const int ri = __builtin_amdgcn_readlane(r, i);
      const int ci = __builtin_amdgcn_readlane(c, i);
      const float vi = __int_as_float(__builtin_amdgcn_readlane(__float_as_int(v), i));

      const float2 kk = *reinterpret_cast<const float2*>(kern + ci * UNITS + f2);
      float* o = out + ri * UNITS + f2;
      // Non-returning f32 adds execute at the (L2-resident) cache side.
      __hip_atomic_fetch_add(o, vi * kk.x, __ATOMIC_RELAXED, __HIP_MEMORY_SCOPE_AGENT);
      __hip_atomic_fetch_add(o + 1, vi * kk.y, __ATOMIC_RELAXED, __HIP_MEMORY_SCOPE_AGENT);
    }
    buf ^= 1;
  }
}

// Epilogue: one wave per 16x16 output tile.
// relu(out + bias) where the +bias broadcast is a rank-1 WMMA:
//   D = A x B + C,  A[m,k] = (k==0),  B[k,n] = (k==0) ? bias[col0+n] : 0
// Only K=0 contributes, so D == C + bias[n] exactly (f32 RNE add).
// A layout (ISA 7.12.2, 32-bit A 16x4): lanes 0-15 hold K=0,1; lanes 16-31 hold K=2,3.
// B's K=0 row lives in VGPR0 lanes 0-15 under either plausible B layout.
__global__ __launch_bounds__(BLOCK_THREADS) void spmm_bias_relu_kernel(
    float* __restrict__ out, const float* __restrict__ bias, int nrows) {
  const int wave_id = blockIdx.x * (BLOCK_THREADS / 32) + (threadIdx.x >> 5);
  const int lane = threadIdx.x & 31;
  const int col_tiles = UNITS / 16;  // 4
  const int tile_r = wave_id / col_tiles;
  const int tile_c = wave_id - tile_r * col_tiles;
  const int row0 = tile_r * 16;
  const int col0 = tile_c * 16;
  if (row0 >= nrows) return;

#if USE_WMMA_EPILOGUE
  if (row0 + 16 <= nrows) {  // whole-wave uniform branch: EXEC stays all-1s for WMMA
    const int half = lane >> 4;  // 0: rows row0..row0+7 / K=0,1 ; 1: rows +8..+15 / K=2,3
    const int ln = lane & 15;
    const int rbase = row0 + half * 8;
    const int col = col0 + ln;

    v8f c;
#pragma unroll
    for (int p = 0; p < 8; ++p) c[p] = out[(rbase + p) * UNITS + col];

    v2f a, b;
    a.x = (half == 0) ? 1.0f : 0.0f;  // A[m,0] = 1 ; A[m,1..3] = 0
    a.y = 0.0f;
    b.x = (half == 0) ? bias[col] : 0.0f;  // B[0,n] = bias ; other rows 0
    b.y = 0.0f;

    c = __builtin_amdgcn_wmma_f32_16x16x4_f32(false, a, false, b, (short)0, c, false, false);

#pragma unroll
    for (int p = 0; p < 8; ++p) out[(rbase + p) * UNITS + col] = fmaxf(c[p], 0.f);
    return;
  }
#endif
  // Scalar path (partial row tile, or WMMA builtin unavailable).
  for (int i = lane; i < 16 * 16; i += 32) {
    const int rr = row0 + (i >> 4);
    const int cc = col0 + (i & 15);
    if (rr < nrows) out[rr * UNITS + cc] = fmaxf(out[rr * UNITS + cc] + bias[cc], 0.f);
  }
}

extern "C" void kernel_launch(void* const* d_in, const int* in_sizes, int n_in,
                              void* d_out, int out_size, void* d_ws, size_t ws_size,
                              hipStream_t stream) {
  (void)n_in;
  (void)d_ws;
  (void)ws_size;

  const int* rows = (const int*)d_in[0];
  const int* cols = (const int*)d_in[1];
  const float* vals = (const float*)d_in[2];
  const float* kern = (const float*)d_in[3];
  const float* bias = (const float*)d_in[4];
  float* out = (float*)d_out;

  const int nnz = in_sizes[0];
  const int total = out_size;  // N * UNITS
  const int nrows = total / UNITS;

  // 1) zero the accumulator (harness poisons d_out)
  const int elem_blocks = (total + 4 * BLOCK_THREADS - 1) / (4 * BLOCK_THREADS);
  spmm_zero_kernel<<<elem_blocks, BLOCK_THREADS, 0, stream>>>(out, total);

  // 2) scatter-add SpMM
  const int num_tiles = (nnz + EDGE_TILE - 1) / EDGE_TILE;
  const int grid = num_tiles < SPMM_MAX_BLOCKS ? num_tiles : SPMM_MAX_BLOCKS;
  if (grid > 0) {
    spmm_atomic_kernel<<<grid, BLOCK_THREADS, 0, stream>>>(rows, cols, vals, kern, out, nnz,
                                                           num_tiles);
  }

  // 3) fused bias + relu epilogue (WMMA rank-1 broadcast-add, one wave per 16x16 tile)
  const int row_tiles = (nrows + 15) / 16;
  const int out_tiles = row_tiles * (UNITS / 16);
  const int waves_per_block = BLOCK_THREADS / 32;
  const int bias_blocks = (out_tiles + waves_per_block - 1) / waves_per_block;
  spmm_bias_relu_kernel<<<bias_blocks, BLOCK_THREADS, 0, stream>>>(out, bias, nrows);
}